// MultiHopGCN_44203803410934
// MI455X (gfx1250) — compile-verified
//
#include <hip/hip_runtime.h>

typedef float v2f __attribute__((ext_vector_type(2)));
typedef float v8f __attribute__((ext_vector_type(8)));

#define FEAT 64
#define TM   32          // output rows per GEMM block
#define NT   256

// ---------------- small utility kernels ----------------

__global__ void k_fill(float* __restrict__ p, float v, int n) {
    int i = blockIdx.x * blockDim.x + threadIdx.x;
    if (i < n) p[i] = v;
}

__global__ void k_deg(const int* __restrict__ row, float* __restrict__ deg, int E) {
    int e = blockIdx.x * blockDim.x + threadIdx.x;
    if (e < E) unsafeAtomicAdd(&deg[row[e]], 1.0f);
}

__global__ void k_rsqrt(const float* __restrict__ deg, float* __restrict__ dis, int n) {
    int i = blockIdx.x * blockDim.x + threadIdx.x;
    if (i < n) dis[i] = rsqrtf(deg[i]);
}

__global__ void k_relu(float* __restrict__ p, int n) {
    int i = blockIdx.x * blockDim.x + threadIdx.x;
    if (i < n) p[i] = fmaxf(p[i], 0.0f);
}

// out_h[node] = dis[node]^2 * h[node]   (self-loop contribution, also inits buffer)
__global__ void k_selfloop(const float* __restrict__ h, const float* __restrict__ dis,
                           float* __restrict__ outh, int n64) {
    int i = blockIdx.x * blockDim.x + threadIdx.x;
    if (i < n64) {
        float d = dis[i >> 6];
        outh[i] = d * d * h[i];
    }
}

// edge scatter: outh[col] += dis[row]*dis[col] * h[row]; 64 lanes per edge (coalesced)
__global__ void k_scatter(const int* __restrict__ row, const int* __restrict__ col,
                          const float* __restrict__ dis, const float* __restrict__ h,
                          float* __restrict__ outh, int E) {
    int t = blockIdx.x * blockDim.x + threadIdx.x;
    int e = t >> 6;
    int f = t & 63;
    if (e < E) {
        int r = row[e], c = col[e];
        float nrm = dis[r] * dis[c];
        unsafeAtomicAdd(&outh[c * FEAT + f], nrm * h[r * FEAT + f]);
    }
}

// acc = (accumulate ? acc : 0) + hw[k] * h
__global__ void k_axpy(float* __restrict__ acc, const float* __restrict__ h,
                       const float* __restrict__ hw, int k, int accumulate, int n64) {
    int i = blockIdx.x * blockDim.x + threadIdx.x;
    if (i < n64) {
        float v = hw[k] * h[i];
        acc[i] = accumulate ? (acc[i] + v) : v;
    }
}

// ---------------- fp32 WMMA GEMM: out[M x 64] = A[M x K] @ W[K x 64] + bias ----------------
// block = 256 threads (8 waves); block computes 32 rows x 64 cols.
// wave w: m-subtile = w>>2 (0..1), n-subtile = w&3 (0..3); each wave = one 16x16 tile.
__global__ __launch_bounds__(256)
void k_gemm_wmma_f32(const float* __restrict__ A, const float* __restrict__ W,
                     const float* __restrict__ bias, float* __restrict__ out, int K) {
    __shared__ float As[TM * 129];   // stride 129 -> conflict-free column reads
    __shared__ float Ws[128 * 65];   // stride 65  -> conflict-free row reads

    const int tid  = threadIdx.x;
    const int row0 = blockIdx.x * TM;

    // stage A tile (32 x K), coalesced
    for (int idx = tid; idx < TM * K; idx += NT) {
        int r = idx / K;
        int c = idx - r * K;
        As[r * (K + 1) + c] = A[(row0 + r) * K + c];
    }
    // stage full W (K x 64), coalesced
    for (int idx = tid; idx < K * FEAT; idx += NT) {
        int r = idx >> 6;
        int c = idx & 63;
        Ws[r * 65 + c] = W[idx];
    }
    __syncthreads();

    const int wave = tid >> 5;
    const int lane = tid & 31;
    const int mtl  = wave >> 2;            // 0..1
    const int ntl  = wave & 3;             // 0..3
    const int lm   = lane & 15;            // A: row m; B/C/D: col n (within tile)
    const int kh   = (lane >> 4) << 1;     // lanes 16-31 hold K+2 / M+8 halves
    const int n    = ntl * 16 + lm;        // column within the 64-wide output

    const float* arow = &As[(mtl * 16 + lm) * (K + 1)];

    v8f acc = {};
    for (int kk = 0; kk < K; kk += 4) {
        v2f a, b;
        // A 16x4 fragment: VGPR r -> K = kk + r + kh
        a.x = arow[kk + kh];
        a.y = arow[kk + kh + 1];
        // B 4x16 fragment: VGPR r -> K = kk + r + kh, N = lane&15
        b.x = Ws[(kk + kh) * 65 + n];
        b.y = Ws[(kk + kh + 1) * 65 + n];
        acc = __builtin_amdgcn_wmma_f32_16x16x4_f32(
                  /*neg_a=*/false, a, /*neg_b=*/false, b,
                  /*c_mod=*/(short)0, acc, /*reuse_a=*/false, /*reuse_b=*/false);
    }

    // C/D layout: VGPR r -> M = r + 8*(lane>=16), N = lane&15
    const float bv    = bias[n];
    const int   mbase = row0 + mtl * 16 + ((lane >> 4) << 3);
#pragma unroll
    for (int r = 0; r < 8; ++r) {
        out[(mbase + r) * FEAT + n] = acc[r] + bv;
    }
}

// ---------------- host-side orchestration ----------------

extern "C" void kernel_launch(void* const* d_in, const int* in_sizes, int n_in,
                              void* d_out, int out_size, void* d_ws, size_t ws_size,
                              hipStream_t stream) {
    const float* x   = (const float*)d_in[0];
    const int*   ei  = (const int*)  d_in[1];
    const float* W1  = (const float*)d_in[2];
    const float* b1  = (const float*)d_in[3];
    const float* hw1 = (const float*)d_in[4];
    const float* W2  = (const float*)d_in[5];
    const float* b2  = (const float*)d_in[6];
    const float* hw2 = (const float*)d_in[7];
    float*       out = (float*)d_out;

    const int N = in_sizes[0] / 128;     // 100000
    const int E = in_sizes[1] / 2;       // 1000000
    const int* row = ei;
    const int* col = ei + E;

    float* ws   = (float*)d_ws;
    float* deg  = ws;                    // N
    float* dis  = ws + N;                // N
    float* bufA = ws + 2 * (size_t)N;    // N*64
    float* bufB = bufA + (size_t)N * FEAT;
    float* acc  = bufB + (size_t)N * FEAT;

    const int n64 = N * FEAT;            // 6.4M
    const int t64 = E * FEAT;            // 64M (fits int)
    const int gN   = (N   + NT - 1) / NT;
    const int gE   = (E   + NT - 1) / NT;
    const int gn64 = (n64 + NT - 1) / NT;
    const int gt64 = (t64 + NT - 1) / NT;
    const int gGemm = N / TM;            // 3125, exact

    // symmetric normalization: deg (incl. self-loop), dis = deg^-1/2
    k_fill <<<gN, NT, 0, stream>>>(deg, 1.0f, N);
    k_deg  <<<gE, NT, 0, stream>>>(row, deg, E);
    k_rsqrt<<<gN, NT, 0, stream>>>(deg, dis, N);

    // ---- layer 1: lin (WMMA) + 2 hops, acc = sum_k hw1[k] * A^(k+1)(lin) ----
    k_gemm_wmma_f32<<<gGemm, NT, 0, stream>>>(x, W1, b1, bufA, 128);
    {
        float* cur = bufA;
        float* nxt = bufB;
        for (int k = 0; k < 2; ++k) {
            k_selfloop<<<gn64, NT, 0, stream>>>(cur, dis, nxt, n64);
            k_scatter <<<gt64, NT, 0, stream>>>(row, col, dis, cur, nxt, E);
            k_axpy    <<<gn64, NT, 0, stream>>>(acc, nxt, hw1, k, k, n64);
            float* t = cur; cur = nxt; nxt = t;
        }
    }
    k_relu<<<gn64, NT, 0, stream>>>(acc, n64);

    // ---- layer 2: lin (WMMA) + 2 hops, result into d_out ----
    k_gemm_wmma_f32<<<gGemm, NT, 0, stream>>>(acc, W2, b2, bufA, 64);
    {
        float* cur = bufA;
        float* nxt = bufB;
        for (int k = 0; k < 2; ++k) {
            k_selfloop<<<gn64, NT, 0, stream>>>(cur, dis, nxt, n64);
            k_scatter <<<gt64, NT, 0, stream>>>(row, col, dis, cur, nxt, E);
            k_axpy    <<<gn64, NT, 0, stream>>>(out, nxt, hw2, k, k, n64);
            float* t = cur; cur = nxt; nxt = t;
        }
    }
}